// SegmentMutualInformationLoss_72481868087475
// MI455X (gfx1250) — compile-verified
//
#include <hip/hip_runtime.h>
#include <math.h>

typedef float v2f __attribute__((ext_vector_type(2)));
typedef float v4f __attribute__((ext_vector_type(4)));
typedef float v8f __attribute__((ext_vector_type(8)));

#define NEGV (-1.0e4f)
#define MAXSP 48   // padded DP size (S=40 -> 3 row tiles of 16)
#define NRT 3      // row tiles (16 rows each)
#define NKC 12     // K chunks of 4 (48/4)

// ---------------- Kernel 1: streaming logsumexp -> lp[row] ----------------
// One block per (batch, span) row of V floats. Online max/sum softmax,
// nontemporal b128 loads (read-once data: don't pollute L2 for other rows).
__global__ __launch_bounds__(256)
void smil_lse_kernel(const float* __restrict__ logits,
                     const int* __restrict__ labels,
                     float* __restrict__ lp, int V, int n_spans) {
  const int row = blockIdx.x;
  const int b = row / n_spans;
  const float* rp = logits + (size_t)row * (size_t)V;
  const v4f* rp4 = reinterpret_cast<const v4f*>(rp);
  const int V4 = V >> 2;

  float m = -3.402823466e38f;
  float s = 0.f;
  for (int i = threadIdx.x; i < V4; i += 256) {
    v4f x = __builtin_nontemporal_load(rp4 + i);
    float mx = fmaxf(fmaxf(x.x, x.y), fmaxf(x.z, x.w));
    float mn = fmaxf(m, mx);
    s = s * __expf(m - mn)
      + __expf(x.x - mn) + __expf(x.y - mn)
      + __expf(x.z - mn) + __expf(x.w - mn);
    m = mn;
  }
  for (int i = (V4 << 2) + threadIdx.x; i < V; i += 256) {
    float x = rp[i];
    float mn = fmaxf(m, x);
    s = s * __expf(m - mn) + __expf(x - mn);
    m = mn;
  }
  // intra-wave (wave32) reduction of (max, scaled-sum)
  #pragma unroll
  for (int off = 16; off > 0; off >>= 1) {
    float mo = __shfl_xor(m, off, 32);
    float so = __shfl_xor(s, off, 32);
    float mn = fmaxf(m, mo);
    s = s * __expf(m - mn) + so * __expf(mo - mn);
    m = mn;
  }
  __shared__ float sm[8], ss[8];
  const int wid = threadIdx.x >> 5;
  if ((threadIdx.x & 31) == 0) { sm[wid] = m; ss[wid] = s; }
  __syncthreads();
  if (threadIdx.x == 0) {
    float M = sm[0], SS = ss[0];
    #pragma unroll
    for (int w = 1; w < 8; ++w) {
      float mo = sm[w], so = ss[w];
      float mn = fmaxf(M, mo);
      SS = SS * __expf(M - mn) + so * __expf(mo - mn);
      M = mn;
    }
    const int label = labels[b];
    float xl = rp[label];
    lp[row] = xl - (M + __logf(SS));
  }
}

// ---------------- Kernel 2: WMMA-based DP scan ----------------
// One wave32 block per batch. I[i] = threshold(M @ I[i-1] + c).
// M (48x48 zero-padded mask matrix) lives in VGPRs as WMMA A-tiles,
// I lives in LDS and feeds the B operand (broadcast across N).
__global__ __launch_bounds__(32)
void smil_dp_kernel(const float* __restrict__ lp,
                    const float* __restrict__ masks,
                    const int* __restrict__ Pp,
                    float* __restrict__ losses,
                    int n_spans, int S) {
  const int b    = blockIdx.x;
  const int lane = threadIdx.x;        // 0..31 (full wave; EXEC all 1s at WMMA)
  const int half = lane >> 4;          // 0: lanes 0-15, 1: lanes 16-31
  const int lrow = lane & 15;
  const int P = Pp[0];

  __shared__ float W[MAXSP * MAXSP];   // mask matrix, zero padded
  __shared__ float cv[MAXSP];          // c[e] = sum_bg LP[e][bg]*M[e][bg]
  __shared__ float I[MAXSP + 16];      // DP state, length S+1 used

  const float* lpb = lp    + (size_t)b * n_spans;
  const float* mkb = masks + (size_t)b * n_spans;

  for (int idx = lane; idx < MAXSP * MAXSP; idx += 32) {
    int e = idx / MAXSP, bg = idx - e * MAXSP;
    float w = 0.f;
    if (e < S && bg <= e) {
      int sid = (e * (e + 1)) / 2 + e - bg;
      w = mkb[sid];
    }
    W[idx] = w;
  }
  for (int e = lane; e < MAXSP; e += 32) {
    float cc = 0.f;
    if (e < S) {
      for (int bg = 0; bg <= e; ++bg) {
        int sid = (e * (e + 1)) / 2 + e - bg;
        cc += lpb[sid] * mkb[sid];
      }
    }
    cv[e] = cc;
  }
  for (int j = lane; j < MAXSP + 16; j += 32)
    I[j] = (j == 0) ? 0.f : NEGV;
  __syncthreads();

  // Preload WMMA A-tiles of W (invariant across DP iterations).
  // 32-bit A 16x4 layout: VGPR0 = K0 (lanes 0-15) / K2 (lanes 16-31),
  //                       VGPR1 = K1 / K3; row M = lane&15.
  v2f A[NRT][NKC];
  #pragma unroll
  for (int rt = 0; rt < NRT; ++rt) {
    int e = rt * 16 + lrow;
    #pragma unroll
    for (int kc = 0; kc < NKC; ++kc) {
      int k0 = kc * 4 + half * 2;
      v2f a;
      a.x = W[e * MAXSP + k0];
      a.y = W[e * MAXSP + k0 + 1];
      A[rt][kc] = a;
    }
  }
  __syncthreads();

  for (int iter = 1; iter <= P; ++iter) {
    v8f acc[NRT];
    #pragma unroll
    for (int rt = 0; rt < NRT; ++rt) { v8f z = {}; acc[rt] = z; }

    #pragma unroll
    for (int kc = 0; kc < NKC; ++kc) {
      // B 4x16 layout mirrors A: VGPR0 = K0/K2, VGPR1 = K1/K3, broadcast over N
      v2f bv;
      bv.x = I[kc * 4 + half * 2];
      bv.y = I[kc * 4 + half * 2 + 1];
      #pragma unroll
      for (int rt = 0; rt < NRT; ++rt) {
        acc[rt] = __builtin_amdgcn_wmma_f32_16x16x4_f32(
            false, A[rt][kc], false, bv, (short)0, acc[rt], false, false);
      }
    }
    __syncthreads();
    // Extract: D VGPR j holds M=j (lanes 0-15) and M=8+j (lanes 16-31);
    // every N column is identical, so lanes 0 and 16 write the results.
    if (lane == 0) I[0] = NEGV;
    if (lane == 0 || lane == 16) {
      #pragma unroll
      for (int rt = 0; rt < NRT; ++rt) {
        int base = rt * 16 + half * 8;
        #pragma unroll
        for (int j = 0; j < 8; ++j) {
          int e = base + j;
          if (e < S) {
            float val = acc[rt][j] + cv[e];
            int jj = e + 1;
            I[jj] = (jj >= iter) ? val : NEGV;
          }
        }
      }
    }
    __syncthreads();
  }
  if (lane == 0) losses[b] = -I[S];
}

// ---------------- Kernel 3: mean over batch ----------------
__global__ void smil_mean_kernel(const float* __restrict__ losses,
                                 float* __restrict__ out, int B) {
  if (threadIdx.x == 0 && blockIdx.x == 0) {
    float s = 0.f;
    for (int i = 0; i < B; ++i) s += losses[i];
    out[0] = s / (float)B;
  }
}

extern "C" void kernel_launch(void* const* d_in, const int* in_sizes, int n_in,
                              void* d_out, int out_size, void* d_ws, size_t ws_size,
                              hipStream_t stream) {
  const float* logits = (const float*)d_in[0];
  const int*   labels = (const int*)d_in[1];
  const float* masks  = (const float*)d_in[2];
  const int*   Pp     = (const int*)d_in[3];
  // d_in[4] = S (also recoverable from n_spans = S(S+1)/2)

  const int B       = in_sizes[1];
  const int rows    = in_sizes[2];          // B * n_spans
  const int n_spans = rows / B;
  const int V       = in_sizes[0] / rows;
  const int S = (int)floor((sqrt(8.0 * (double)n_spans + 1.0) - 1.0) / 2.0 + 0.5);

  float* lp     = (float*)d_ws;             // rows floats
  float* losses = lp + rows;                // B floats

  smil_lse_kernel<<<rows, 256, 0, stream>>>(logits, labels, lp, V, n_spans);
  smil_dp_kernel<<<B, 32, 0, stream>>>(lp, masks, Pp, losses, n_spans, S);
  smil_mean_kernel<<<1, 32, 0, stream>>>(losses, (float*)d_out, B);
}